// MPCSolver_73392401154413
// MI455X (gfx1250) — compile-verified
//
#include <hip/hip_runtime.h>

// ============================================================================
// MPC solver (Adam over RK4 rollout of two 5->256->5 tanh MLPs), MI455X.
//
// Strategy: the problem is a long serial chain of tiny GEMMs (B=1024, K/N=5 or
// 256). Total FLOPs ~1e11, working set < 600KB -> latency bound, not roofline
// bound. One persistent workgroup (1024 threads = 32 wave32 waves on one WGP)
// holds all weights + RK4 state + cotangents in LDS (~275KB of the 320KB),
// runs all 100 Adam iterations internally, and uses native f32 WMMA
// (v_wmma_f32_16x16x4_f32) so precision matches the f32 reference exactly.
// Activations use the CDNA5 native V_TANH_F32 transcendental when available
// (TRANS ops co-execute with the WMMA pipe).
// ============================================================================

#define HORIZON   10
#define DT        0.01f
#define LR        0.1f
#define BETA1_C   0.9f
#define BETA2_C   0.999f
#define EPS_C     1e-8f
#define COST_COEF 0.01f
#define INV_COEF  0.1f
#define SDIM      5
#define HID       256
#define BATCH     1024

#define NTHREADS  1024
#define TSTRIDE   17            // 16x16 transpose tile, padded stride
#define TSIZE     (16*TSTRIDE)  // 272 floats per wave

typedef __attribute__((ext_vector_type(2))) float v2f;
typedef __attribute__((ext_vector_type(8))) float v8f;

// ---------------- LDS layout (float offsets) ----------------
constexpr int OFF_WF1  = 0;                      // 5x256
constexpr int OFF_WG1  = OFF_WF1 + SDIM*HID;
constexpr int OFF_WF2  = OFF_WG1 + SDIM*HID;     // 256x5
constexpr int OFF_WG2  = OFF_WF2 + HID*SDIM;
constexpr int OFF_BF1  = OFF_WG2 + HID*SDIM;     // 256
constexpr int OFF_BG1  = OFF_BF1 + HID;
constexpr int OFF_BF2  = OFF_BG1 + HID;          // 16 (padded)
constexpr int OFF_BG2  = OFF_BF2 + 16;
constexpr int OFF_UL   = OFF_BG2 + 16;           // u_t staged, 1024
constexpr int OFF_DPL  = OFF_UL  + BATCH;        // dp = k1[:,3], 1024
constexpr int OFF_RED  = OFF_DPL + BATCH;        // loss reduction, 1024
constexpr int OFF_LOSS = OFF_RED + BATCH;        // 4
constexpr int OFF_XS1  = OFF_LOSS + 4;           // state buffers, BATCH*SDIM each
constexpr int OFF_XS2  = OFF_XS1 + BATCH*SDIM;
constexpr int OFF_XS3  = OFF_XS2 + BATCH*SDIM;
constexpr int OFF_XS4  = OFF_XS3 + BATCH*SDIM;
constexpr int OFF_KB   = OFF_XS4 + BATCH*SDIM;
constexpr int OFF_XBAR = OFF_KB  + BATCH*SDIM;
constexpr int OFF_CBAR = OFF_XBAR+ BATCH*SDIM;
constexpr int OFF_KBR  = OFF_CBAR+ BATCH*SDIM;
constexpr int OFF_GST  = OFF_KBR + BATCH*SDIM;
constexpr int OFF_GOUT = OFF_GST + BATCH*SDIM;
constexpr int OFF_TRANS= OFF_GOUT+ BATCH*SDIM;   // 32 waves * 272
constexpr int SMEM_FLOATS = OFF_TRANS + 32*TSIZE;
constexpr int SMEM_BYTES  = SMEM_FLOATS * 4;
static_assert(SMEM_BYTES <= 320*1024, "exceeds WGP LDS");

// ---------------- CDNA5 native tanh (V_TANH_F32) if declared ----------------
__device__ __forceinline__ float fast_tanh(float x) {
#if __has_builtin(__builtin_amdgcn_tanhf)
  return __builtin_amdgcn_tanhf(x);
#elif __has_builtin(__builtin_amdgcn_tanh_f32)
  return __builtin_amdgcn_tanh_f32(x);
#else
  return tanhf(x);
#endif
}

// ---------------- CDNA5 f32 WMMA, D = A(16x4) * B(4x16) + C ----------------
__device__ __forceinline__ v8f wmma4(v2f a, v2f b, v8f c) {
  return __builtin_amdgcn_wmma_f32_16x16x4_f32(
      /*neg_a=*/false, a, /*neg_b=*/false, b,
      /*c_mod=*/(short)0, c, /*reuse_a=*/false, /*reuse_b=*/false);
}

// A-frag from a BATCHx5 row-major LDS buffer; K padded to 8 with zeros.
// Layout: lane l -> row (row0 + (l&15)); vgpr j -> k = k0 + 2*(l>=16) + j.
// Optional per-row scale (uscale) implements (kbar * u) without a buffer.
__device__ __forceinline__ v2f a_state(const float* buf, const float* uscale,
                                       int row0, int k0, int lane) {
  int m  = row0 + (lane & 15);
  int kb = k0 + ((lane >> 4) << 1);
  int kc = kb + 1;
  float s  = uscale ? uscale[m] : 1.0f;
  float x0 = buf[m*SDIM + (kb < SDIM ? kb : 0)];
  float x1 = buf[m*SDIM + (kc < SDIM ? kc : 0)];
  v2f a;
  a.x = (kb < SDIM) ? x0 * s : 0.0f;
  a.y = (kc < SDIM) ? x1 * s : 0.0f;
  return a;
}

// B-frag from W1 (5x256 row-major) rows k0..k0+3 (zero pad past 5), cols n0..n0+15.
__device__ __forceinline__ v2f b_w1(const float* W1, int k0, int n0, int lane) {
  int n  = n0 + (lane & 15);
  int kb = k0 + ((lane >> 4) << 1);
  int kc = kb + 1;
  float w0 = W1[(kb < SDIM ? kb : 0)*HID + n];
  float w1 = W1[(kc < SDIM ? kc : 0)*HID + n];
  v2f b;
  b.x = (kb < SDIM) ? w0 : 0.0f;
  b.y = (kc < SDIM) ? w1 : 0.0f;
  return b;
}

// B-frag from W2 (256x5 row-major), rows k0..k0+3 (hidden), cols padded to 16.
__device__ __forceinline__ v2f b_w2(const float* W2, int k0, int lane) {
  int n  = lane & 15;
  int kb = k0 + ((lane >> 4) << 1);
  int nc = (n < SDIM) ? n : 0;
  float w0 = W2[kb*SDIM + nc];
  float w1 = W2[(kb + 1)*SDIM + nc];
  v2f b;
  b.x = (n < SDIM) ? w0 : 0.0f;
  b.y = (n < SDIM) ? w1 : 0.0f;
  return b;
}

// B-frag of W2^T (5x256): element (k,n) = W2[n*5+k]; rows padded past 5.
__device__ __forceinline__ v2f b_w2t(const float* W2, int k0, int n0, int lane) {
  int n  = n0 + (lane & 15);
  int kb = k0 + ((lane >> 4) << 1);
  int kc = kb + 1;
  float w0 = W2[n*SDIM + (kb < SDIM ? kb : 0)];
  float w1 = W2[n*SDIM + (kc < SDIM ? kc : 0)];
  v2f b;
  b.x = (kb < SDIM) ? w0 : 0.0f;
  b.y = (kc < SDIM) ? w1 : 0.0f;
  return b;
}

// B-frag of W1^T (256x5): element (k,n) = W1[n*256+k]; cols padded past 5.
__device__ __forceinline__ v2f b_w1t(const float* W1, int k0, int lane) {
  int n  = lane & 15;
  int kb = k0 + ((lane >> 4) << 1);
  int nc = (n < SDIM) ? n : 0;
  float w0 = W1[nc*HID + kb];
  float w1 = W1[nc*HID + kb + 1];
  v2f b;
  b.x = (n < SDIM) ? w0 : 0.0f;
  b.y = (n < SDIM) ? w1 : 0.0f;
  return b;
}

// A-frag from the wave-private 16x16 transpose tile (row-major, stride 17).
__device__ __forceinline__ v2f a_trans(const float* tb, int k0, int lane) {
  int m  = lane & 15;
  int kb = k0 + ((lane >> 4) << 1);
  v2f a;
  a.x = tb[m*TSTRIDE + kb];
  a.y = tb[m*TSTRIDE + kb + 1];
  return a;
}

// Store a 16x16 C/D tile (C layout: lane=N, vgpr r = row r / r+8) row-major.
__device__ __forceinline__ void store_c_tile(float* tb, v8f c, int lane) {
  int col = lane & 15;
  int rb  = (lane < 16) ? 0 : 8;
#pragma unroll
  for (int r = 0; r < 8; ++r) tb[(rb + r)*TSTRIDE + col] = c[r];
}

// Store valid cols (0..4) of a C tile into a BATCHx5 row-major LDS buffer.
template <bool ACC>
__device__ __forceinline__ void store_c_state(float* buf, int row0, v8f c, int lane) {
  int col = lane & 15;
  int rb  = (lane < 16) ? 0 : 8;
  if (col < SDIM) {
#pragma unroll
    for (int r = 0; r < 8; ++r) {
      int idx = (row0 + rb + r)*SDIM + col;
      if (ACC) buf[idx] += c[r]; else buf[idx] = c[r];
    }
  }
}

// One 16-row tile of MLP(x) = tanh(x@W1 + b1) @ W2 + b2, fused via LDS transpose.
__device__ __forceinline__ v8f mlp_forward_tile(const float* xs, int row0,
    const float* W1, const float* b1, const float* W2, const float* b2,
    float* tb, int lane) {
  v8f acc2 = {};
#pragma unroll 1
  for (int n = 0; n < 16; ++n) {
    v8f h = {};
    h = wmma4(a_state(xs, nullptr, row0, 0, lane), b_w1(W1, 0, n*16, lane), h);
    h = wmma4(a_state(xs, nullptr, row0, 4, lane), b_w1(W1, 4, n*16, lane), h);
    float bias = b1[n*16 + (lane & 15)];
#pragma unroll
    for (int r = 0; r < 8; ++r) h[r] = fast_tanh(h[r] + bias);
    store_c_tile(tb, h, lane);                 // C-layout -> row-major (wave-private)
#pragma unroll
    for (int ks = 0; ks < 4; ++ks)
      acc2 = wmma4(a_trans(tb, ks*4, lane), b_w2(W2, n*16 + ks*4, lane), acc2);
  }
  float b2v = ((lane & 15) < SDIM) ? b2[lane & 15] : 0.0f;
#pragma unroll
  for (int r = 0; r < 8; ++r) acc2[r] += b2v;
  return acc2;
}

// K = f(x) + g(x) * u for this wave's 32 rows.
__device__ __forceinline__ void system_eval(
    const float* xs, float* K, const float* uL,
    const float* Wf1, const float* bf1, const float* Wf2, const float* bf2,
    const float* Wg1, const float* bg1, const float* Wg2, const float* bg2,
    float* tb, int lane, int wave) {
#pragma unroll
  for (int mt = 0; mt < 2; ++mt) {
    int row0 = (wave*2 + mt)*16;
    v8f f = mlp_forward_tile(xs, row0, Wf1, bf1, Wf2, bf2, tb, lane);
    v8f g = mlp_forward_tile(xs, row0, Wg1, bg1, Wg2, bg2, tb, lane);
    int col = lane & 15;
    int rb  = (lane < 16) ? 0 : 8;
    if (col < SDIM) {
#pragma unroll
      for (int r = 0; r < 8; ++r) {
        int row = row0 + rb + r;
        K[row*SDIM + col] = f[r] + g[r]*uL[row];
      }
    }
  }
}

// VJP of system() at xs with output cotangent kbar:
//   gst  = d(kbar . sys)/dx   (f path written, g path accumulated)
//   gout = g(xs) = tanh(xs@Wg1+bg1)@Wg2 + bg2   (for du = sum kbar*gout)
__device__ __forceinline__ void vjp_stage(
    const float* xs, const float* kbar, const float* uL,
    const float* Wf1, const float* bf1, const float* Wf2,
    const float* Wg1, const float* bg1, const float* Wg2, const float* bg2,
    float* gst, float* gout, float* tb, int lane, int wave) {
#pragma unroll
  for (int mt = 0; mt < 2; ++mt) {
    int row0 = (wave*2 + mt)*16;

    // ---- f path: ((kbar @ Wf2^T) * tanh') @ Wf1^T ----
    v8f gx = {};
#pragma unroll 1
    for (int n = 0; n < 16; ++n) {
      v8f h = {};
      h = wmma4(a_state(xs, nullptr, row0, 0, lane), b_w1(Wf1, 0, n*16, lane), h);
      h = wmma4(a_state(xs, nullptr, row0, 4, lane), b_w1(Wf1, 4, n*16, lane), h);
      float bias = bf1[n*16 + (lane & 15)];
#pragma unroll
      for (int r = 0; r < 8; ++r) h[r] = fast_tanh(h[r] + bias);
      v8f t1 = {};
      t1 = wmma4(a_state(kbar, nullptr, row0, 0, lane), b_w2t(Wf2, 0, n*16, lane), t1);
      t1 = wmma4(a_state(kbar, nullptr, row0, 4, lane), b_w2t(Wf2, 4, n*16, lane), t1);
#pragma unroll
      for (int r = 0; r < 8; ++r) t1[r] *= (1.0f - h[r]*h[r]);
      store_c_tile(tb, t1, lane);
#pragma unroll
      for (int ks = 0; ks < 4; ++ks)
        gx = wmma4(a_trans(tb, ks*4, lane), b_w1t(Wf1, n*16 + ks*4, lane), gx);
    }
    store_c_state<false>(gst, row0, gx, lane);

    // ---- g path: (((kbar*u) @ Wg2^T) * tanh') @ Wg1^T ; also gout = g(x) ----
    v8f gx2 = {};
    v8f go  = {};
#pragma unroll 1
    for (int n = 0; n < 16; ++n) {
      v8f h = {};
      h = wmma4(a_state(xs, nullptr, row0, 0, lane), b_w1(Wg1, 0, n*16, lane), h);
      h = wmma4(a_state(xs, nullptr, row0, 4, lane), b_w1(Wg1, 4, n*16, lane), h);
      float bias = bg1[n*16 + (lane & 15)];
#pragma unroll
      for (int r = 0; r < 8; ++r) h[r] = fast_tanh(h[r] + bias);
      store_c_tile(tb, h, lane);                 // transpose h for gout GEMM
#pragma unroll
      for (int ks = 0; ks < 4; ++ks)
        go = wmma4(a_trans(tb, ks*4, lane), b_w2(Wg2, n*16 + ks*4, lane), go);
      v8f t1 = {};
      t1 = wmma4(a_state(kbar, uL, row0, 0, lane), b_w2t(Wg2, 0, n*16, lane), t1);
      t1 = wmma4(a_state(kbar, uL, row0, 4, lane), b_w2t(Wg2, 4, n*16, lane), t1);
#pragma unroll
      for (int r = 0; r < 8; ++r) t1[r] *= (1.0f - h[r]*h[r]);
      store_c_tile(tb, t1, lane);
#pragma unroll
      for (int ks = 0; ks < 4; ++ks)
        gx2 = wmma4(a_trans(tb, ks*4, lane), b_w1t(Wg1, n*16 + ks*4, lane), gx2);
    }
    store_c_state<true>(gst, row0, gx2, lane);
    float b2v = ((lane & 15) < SDIM) ? bg2[lane & 15] : 0.0f;
#pragma unroll
    for (int r = 0; r < 8; ++r) go[r] += b2v;
    store_c_state<false>(gout, row0, go, lane);
  }
}

// ============================================================================
extern "C" __global__ __launch_bounds__(NTHREADS, 1)
void mpc_solver_kernel(const float* __restrict__ gx0,
                       const float* __restrict__ gWf1, const float* __restrict__ gbf1,
                       const float* __restrict__ gWf2, const float* __restrict__ gbf2,
                       const float* __restrict__ gWg1, const float* __restrict__ gbg1,
                       const float* __restrict__ gWg2, const float* __restrict__ gbg2,
                       const int*   __restrict__ pIters,
                       float* __restrict__ out, float* __restrict__ ws) {
  extern __shared__ float sm[];
  const int tid  = (int)threadIdx.x;
  const int lane = tid & 31;
  const int wave = tid >> 5;
  const int row  = tid;                       // one batch row per thread

  float* Wf1 = sm + OFF_WF1;  float* Wg1 = sm + OFF_WG1;
  float* Wf2 = sm + OFF_WF2;  float* Wg2 = sm + OFF_WG2;
  float* bf1 = sm + OFF_BF1;  float* bg1 = sm + OFF_BG1;
  float* bf2 = sm + OFF_BF2;  float* bg2 = sm + OFF_BG2;
  float* uL  = sm + OFF_UL;   float* dpL = sm + OFF_DPL;
  float* red = sm + OFF_RED;  float* lossL = sm + OFF_LOSS;
  float* XS1 = sm + OFF_XS1;  float* XS2 = sm + OFF_XS2;  // XS2 doubles as fwd stage
  float* XS3 = sm + OFF_XS3;  float* XS4 = sm + OFF_XS4;  // XS3 doubles as fwd accum
  float* KB  = sm + OFF_KB;
  float* XBAR= sm + OFF_XBAR; float* CBAR= sm + OFF_CBAR;
  float* KBR = sm + OFF_KBR;  float* GST = sm + OFF_GST;
  float* GOUT= sm + OFF_GOUT;
  float* tb  = sm + OFF_TRANS + wave*TSIZE;

  // global workspace: u, m, v (Adam), saved trajectory x_t
  float* uW   = ws;
  float* mW   = ws + HORIZON*BATCH;
  float* vW   = ws + 2*HORIZON*BATCH;
  float* traj = ws + 3*HORIZON*BATCH;   // HORIZON * BATCH * SDIM

  // ---- init: weights -> LDS, Adam state -> 0 ----
  for (int i = tid; i < SDIM*HID; i += NTHREADS) { Wf1[i] = gWf1[i]; Wg1[i] = gWg1[i]; }
  for (int i = tid; i < HID*SDIM; i += NTHREADS) { Wf2[i] = gWf2[i]; Wg2[i] = gWg2[i]; }
  for (int i = tid; i < HID;      i += NTHREADS) { bf1[i] = gbf1[i]; bg1[i] = gbg1[i]; }
  if (tid < 16) { bf2[tid] = (tid < SDIM) ? gbf2[tid] : 0.0f;
                  bg2[tid] = (tid < SDIM) ? gbg2[tid] : 0.0f; }
  for (int i = tid; i < HORIZON*BATCH; i += NTHREADS) { uW[i] = 0.f; mW[i] = 0.f; vW[i] = 0.f; }
  __syncthreads();

  const int ITERS = pIters[0];
  float b1pow = 1.0f, b2pow = 1.0f;

#pragma unroll 1
  for (int it = 0; it < ITERS; ++it) {
    b1pow *= BETA1_C;  b2pow *= BETA2_C;

#pragma unroll
    for (int c = 0; c < SDIM; ++c) XS1[row*SDIM + c] = gx0[row*SDIM + c];
    float lossPart = 0.0f;
    __syncthreads();

    // ======================= forward (RK4 scan + loss) =======================
#pragma unroll 1
    for (int t = 0; t < HORIZON; ++t) {
      uL[row] = uW[t*BATCH + row];
#pragma unroll
      for (int c = 0; c < SDIM; ++c) traj[(t*BATCH + row)*SDIM + c] = XS1[row*SDIM + c];
      __syncthreads();

      system_eval(XS1, KB, uL, Wf1,bf1,Wf2,bf2, Wg1,bg1,Wg2,bg2, tb, lane, wave); // k1
      __syncthreads();
      {
        float inv = XS1[row*SDIM + 4];
        float dp  = KB [row*SDIM + 3];
        float uv  = uL[row];
        lossPart += -inv*dp + COST_COEF*DT*uv*uv + INV_COEF*DT*inv*inv;
#pragma unroll
        for (int c = 0; c < SDIM; ++c) {
          float x = XS1[row*SDIM + c], k = KB[row*SDIM + c];
          XS3[row*SDIM + c] = x + (DT/6.0f)*k;     // accumulator
          XS2[row*SDIM + c] = x + 0.5f*DT*k;       // next stage
        }
      }
      __syncthreads();
      system_eval(XS2, KB, uL, Wf1,bf1,Wf2,bf2, Wg1,bg1,Wg2,bg2, tb, lane, wave); // k2
      __syncthreads();
#pragma unroll
      for (int c = 0; c < SDIM; ++c) {
        float k = KB[row*SDIM + c];
        XS3[row*SDIM + c] += (DT/3.0f)*k;
        XS2[row*SDIM + c]  = XS1[row*SDIM + c] + 0.5f*DT*k;
      }
      __syncthreads();
      system_eval(XS2, KB, uL, Wf1,bf1,Wf2,bf2, Wg1,bg1,Wg2,bg2, tb, lane, wave); // k3
      __syncthreads();
#pragma unroll
      for (int c = 0; c < SDIM; ++c) {
        float k = KB[row*SDIM + c];
        XS3[row*SDIM + c] += (DT/3.0f)*k;
        XS2[row*SDIM + c]  = XS1[row*SDIM + c] + DT*k;
      }
      __syncthreads();
      system_eval(XS2, KB, uL, Wf1,bf1,Wf2,bf2, Wg1,bg1,Wg2,bg2, tb, lane, wave); // k4
      __syncthreads();
#pragma unroll
      for (int c = 0; c < SDIM; ++c)
        XS1[row*SDIM + c] = XS3[row*SDIM + c] + (DT/6.0f)*KB[row*SDIM + c];
      __syncthreads();
    }

    // deterministic tree reduction of the loss
    red[tid] = lossPart;
    __syncthreads();
#pragma unroll 1
    for (int s = NTHREADS/2; s > 0; s >>= 1) {
      if (tid < s) red[tid] += red[tid + s];
      __syncthreads();
    }
    if (tid == 0) lossL[0] = red[0] / (float)BATCH;

    // ======================= backward (reverse scan) =======================
#pragma unroll
    for (int c = 0; c < SDIM; ++c) CBAR[row*SDIM + c] = 0.0f;
    __syncthreads();

#pragma unroll 1
    for (int t = HORIZON - 1; t >= 0; --t) {
      // hide the only global latency: prefetch next step's trajectory + u
      if (t > 0) {
        __builtin_prefetch(&traj[((t - 1)*BATCH + row)*SDIM], 0, 0);
        __builtin_prefetch(&uW[(t - 1)*BATCH + row], 0, 0);
      }
      uL[row] = uW[t*BATCH + row];
#pragma unroll
      for (int c = 0; c < SDIM; ++c) XS1[row*SDIM + c] = traj[(t*BATCH + row)*SDIM + c];
      __syncthreads();

      // recompute stage states x1..x4
      system_eval(XS1, KB, uL, Wf1,bf1,Wf2,bf2, Wg1,bg1,Wg2,bg2, tb, lane, wave);
      __syncthreads();
      dpL[row] = KB[row*SDIM + 3];
#pragma unroll
      for (int c = 0; c < SDIM; ++c) XS2[row*SDIM + c] = XS1[row*SDIM + c] + 0.5f*DT*KB[row*SDIM + c];
      __syncthreads();
      system_eval(XS2, KB, uL, Wf1,bf1,Wf2,bf2, Wg1,bg1,Wg2,bg2, tb, lane, wave);
      __syncthreads();
#pragma unroll
      for (int c = 0; c < SDIM; ++c) XS3[row*SDIM + c] = XS1[row*SDIM + c] + 0.5f*DT*KB[row*SDIM + c];
      __syncthreads();
      system_eval(XS3, KB, uL, Wf1,bf1,Wf2,bf2, Wg1,bg1,Wg2,bg2, tb, lane, wave);
      __syncthreads();
#pragma unroll
      for (int c = 0; c < SDIM; ++c) XS4[row*SDIM + c] = XS1[row*SDIM + c] + DT*KB[row*SDIM + c];
      __syncthreads();

      float ub = 0.0f;
      // stage 4: kbar4 = (DT/6) c
#pragma unroll
      for (int c = 0; c < SDIM; ++c) {
        float cb = CBAR[row*SDIM + c];
        XBAR[row*SDIM + c] = cb;
        KBR [row*SDIM + c] = (DT/6.0f)*cb;
      }
      __syncthreads();
      vjp_stage(XS4, KBR, uL, Wf1,bf1,Wf2, Wg1,bg1,Wg2,bg2, GST, GOUT, tb, lane, wave);
      __syncthreads();
#pragma unroll
      for (int c = 0; c < SDIM; ++c) {
        float kb = KBR[row*SDIM + c];
        ub += kb * GOUT[row*SDIM + c];
        float g4 = GST[row*SDIM + c];
        XBAR[row*SDIM + c] += g4;
        KBR [row*SDIM + c]  = (DT/3.0f)*CBAR[row*SDIM + c] + DT*g4;    // kbar3
      }
      __syncthreads();
      vjp_stage(XS3, KBR, uL, Wf1,bf1,Wf2, Wg1,bg1,Wg2,bg2, GST, GOUT, tb, lane, wave);
      __syncthreads();
#pragma unroll
      for (int c = 0; c < SDIM; ++c) {
        float kb = KBR[row*SDIM + c];
        ub += kb * GOUT[row*SDIM + c];
        float g3 = GST[row*SDIM + c];
        XBAR[row*SDIM + c] += g3;
        KBR [row*SDIM + c]  = (DT/3.0f)*CBAR[row*SDIM + c] + 0.5f*DT*g3; // kbar2
      }
      __syncthreads();
      vjp_stage(XS2, KBR, uL, Wf1,bf1,Wf2, Wg1,bg1,Wg2,bg2, GST, GOUT, tb, lane, wave);
      __syncthreads();
#pragma unroll
      for (int c = 0; c < SDIM; ++c) {
        float kb = KBR[row*SDIM + c];
        ub += kb * GOUT[row*SDIM + c];
        float g2 = GST[row*SDIM + c];
        XBAR[row*SDIM + c] += g2;
        KBR [row*SDIM + c]  = (DT/6.0f)*CBAR[row*SDIM + c] + 0.5f*DT*g2; // kbar1
      }
      KBR[row*SDIM + 3] += -XS1[row*SDIM + 4] / (float)BATCH;  // dp = k1[:,3] seed
      __syncthreads();
      vjp_stage(XS1, KBR, uL, Wf1,bf1,Wf2, Wg1,bg1,Wg2,bg2, GST, GOUT, tb, lane, wave);
      __syncthreads();
#pragma unroll
      for (int c = 0; c < SDIM; ++c) {
        ub += KBR[row*SDIM + c] * GOUT[row*SDIM + c];
        XBAR[row*SDIM + c] += GST[row*SDIM + c];
      }
      {
        float inv = XS1[row*SDIM + 4];                         // inv = x_t[:,4] seed
        XBAR[row*SDIM + 4] += (-dpL[row] + 2.0f*INV_COEF*DT*inv) / (float)BATCH;
      }
#pragma unroll
      for (int c = 0; c < SDIM; ++c) CBAR[row*SDIM + c] = XBAR[row*SDIM + c];

      // grad_u[t] + inline Adam update
      {
        float g  = ub + 2.0f*COST_COEF*DT*uL[row] / (float)BATCH;
        int idx  = t*BATCH + row;
        float mm = BETA1_C*mW[idx] + (1.0f - BETA1_C)*g;
        float vv = BETA2_C*vW[idx] + (1.0f - BETA2_C)*g*g;
        mW[idx] = mm;  vW[idx] = vv;
        float mh = mm / (1.0f - b1pow);
        float vh = vv / (1.0f - b2pow);
        uW[idx] -= LR * mh / (sqrtf(vh) + EPS_C);
      }
      __syncthreads();
    }
  }

  // outputs: u (HORIZON*BATCH floats) then final loss
  for (int i = tid; i < HORIZON*BATCH; i += NTHREADS) out[i] = uW[i];
  if (tid == 0) out[HORIZON*BATCH] = lossL[0];
}

// ============================================================================
extern "C" void kernel_launch(void* const* d_in, const int* in_sizes, int n_in,
                              void* d_out, int out_size, void* d_ws, size_t ws_size,
                              hipStream_t stream) {
  (void)in_sizes; (void)n_in; (void)out_size; (void)ws_size;
  const float* x0  = (const float*)d_in[0];
  const float* Wf1 = (const float*)d_in[1];
  const float* bf1 = (const float*)d_in[2];
  const float* Wf2 = (const float*)d_in[3];
  const float* bf2 = (const float*)d_in[4];
  const float* Wg1 = (const float*)d_in[5];
  const float* bg1 = (const float*)d_in[6];
  const float* Wg2 = (const float*)d_in[7];
  const float* bg2 = (const float*)d_in[8];
  const int* iters = (const int*)d_in[9];
  float* out = (float*)d_out;
  float* ws  = (float*)d_ws;

  (void)hipFuncSetAttribute((const void*)mpc_solver_kernel,
                            hipFuncAttributeMaxDynamicSharedMemorySize, SMEM_BYTES);
  mpc_solver_kernel<<<1, NTHREADS, SMEM_BYTES, stream>>>(
      x0, Wf1, bf1, Wf2, bf2, Wg1, bg1, Wg2, bg2, iters, out, ws);
}